// DADMM_54640573940346
// MI455X (gfx1250) — compile-verified
//
#include <hip/hip_runtime.h>

// D-ADMM, fully unrolled in one persistent kernel.
// Grid: 1024 blocks x 256 threads (8 wave32s); wave = (agent p, local row r).
// Residency: X, mu, own-a in VGPRs (float4 chunks); a (neighbor copy), omega in LDS.
// a0 arrives via TDM tensor_load_to_lds; final a leaves via tensor_store_from_lds.
// All global/LDS data movement is b128.

static constexpr int PA    = 4;     // agents
static constexpr int DD    = 784;   // features
static constexpr int NN    = 2048;  // samples
static constexpr int NQ    = 196;   // float4 per row (784/4)
static constexpr int NC4   = 7;     // ceil(196/32) float4 chunks per lane
static constexpr int ROWS  = 2;     // samples per workgroup
static constexpr int BLOCK = 256;   // 8 waves

typedef unsigned int u32x4 __attribute__((ext_vector_type(4)));
typedef int          i32x8 __attribute__((ext_vector_type(8)));
typedef int          i32x4 __attribute__((ext_vector_type(4)));

// TDM D# for a 2D tile: ROWS lines of DD f32, line stride DD elements (ISA §8.3/8.4).
__device__ __forceinline__ void tdm_desc(unsigned long long gaddr, unsigned lds_off,
                                         u32x4& g0, i32x8& g1) {
  g0.x = 1u;                                        // count=1 (valid user descriptor)
  g0.y = lds_off;                                   // lds_addr (bytes)
  g0.z = (unsigned)(gaddr & 0xFFFFFFFFull);         // global_addr[31:0]
  g0.w = (unsigned)(gaddr >> 32) | 0x80000000u;     // global_addr[56:32] | type=2 @ [127:126]
  g1[0] = (2 << 16);                                // wg_mask=0 | data_size=2 (4B) @ [17:16]
  g1[1] = (int)((DD & 0xFFFF) << 16);               // tensor_dim0 lo16 @ [63:48]
  g1[2] = (int)((DD >> 16) | (ROWS << 16));         // tensor_dim0 hi16 | tensor_dim1 lo16
  g1[3] = (int)(DD << 16);                          // tensor_dim1 hi16(0) | tile_dim0 @ [127:112]
  g1[4] = ROWS;                                     // tile_dim1=2 | tile_dim2=0
  g1[5] = DD;                                       // tensor_dim0_stride[31:0] (elements)
  g1[6] = 0;
  g1[7] = 0;
}

__global__ __launch_bounds__(BLOCK)
void DADMM_54640573940346_kernel(const float* __restrict__ X,     // [P,N,D]
                                 const float* __restrict__ y,     // [P,N]
                                 const float* __restrict__ a0,    // [P,N,D]
                                 const float* __restrict__ om0,   // [P,N]
                                 const float* __restrict__ mu0,   // [P,N,D]
                                 const float* __restrict__ lam0,  // [P,N]
                                 const float* __restrict__ hyp,   // [K,P,5]
                                 const int*   __restrict__ maxIter,
                                 float*       __restrict__ out)   // a flat, then omega
{
  __shared__ __align__(16) float a_lds[PA * ROWS][DD];   // [p*ROWS + r][d]  (~25 KB)
  __shared__ float om_lds[PA * ROWS];

  const int tid  = threadIdx.x;
  const int lane = tid & 31;
  const int w    = tid >> 5;               // 0..7
  const int p    = w >> 1;                 // agent
  const int r    = w & 1;                  // local row
  const int n    = blockIdx.x * ROWS + r;
  const size_t rowoff = ((size_t)p * NN + n) * DD;
  const int pn = p * NN + n;

  // ---- TDM: async-load a0 tiles (2 rows x 784 f32 per agent) into LDS ----
  if (w == 0) {
    const unsigned lds_base = (unsigned)(size_t)(&a_lds[0][0]);
    #pragma unroll
    for (int pp = 0; pp < PA; ++pp) {
      u32x4 g0; i32x8 g1;
      i32x4 z4 = {0, 0, 0, 0};
      i32x8 z8 = {0, 0, 0, 0, 0, 0, 0, 0};
      unsigned long long ga =
          (unsigned long long)(size_t)(a0 + ((size_t)pp * NN + (size_t)blockIdx.x * ROWS) * DD);
      tdm_desc(ga, lds_base + (unsigned)(pp * ROWS * DD * 4), g0, g1);
      __builtin_amdgcn_tensor_load_to_lds(g0, g1, z4, z4, z8, 0);
    }
    __builtin_amdgcn_s_wait_tensorcnt(0);
  }

  // ---- register-resident per-lane state, b128 chunks (q = c*32 + lane) ----
  const float4* __restrict__ Xv = (const float4*)(X + rowoff);
  const float4* __restrict__ Mv = (const float4*)(mu0 + rowoff);
  float4 Xr[NC4], mur[NC4], ar[NC4], nbr[NC4];
  const float4 f4z = make_float4(0.f, 0.f, 0.f, 0.f);
  #pragma unroll
  for (int c = 0; c < NC4; ++c) {
    const int q = c * 32 + lane;
    const bool ok = q < NQ;
    Xr[c]  = ok ? Xv[q] : f4z;
    mur[c] = ok ? Mv[q] : f4z;
    nbr[c] = f4z;
  }
  float lamv = lam0[pn];
  const float yv = y[pn];
  if (lane == 0) om_lds[w] = om0[pn];
  const int K = maxIter[0];

  __syncthreads();   // a0 in LDS + omega0 visible

  // 4-cycle graph: neighbors of even agents = {1,3}; of odd agents = {0,2}
  const int nb0 = ((p & 1) == 0) ? 1 : 0;
  const int nb1 = ((p & 1) == 0) ? 3 : 2;
  float4* __restrict__ aself4 = (float4*)&a_lds[w][0];
  const float4* __restrict__ anb0_4 = (const float4*)&a_lds[nb0 * ROWS + r][0];
  const float4* __restrict__ anb1_4 = (const float4*)&a_lds[nb1 * ROWS + r][0];

  // pull own a row into registers (stays there for all iterations)
  #pragma unroll
  for (int c = 0; c < NC4; ++c) {
    const int q = c * 32 + lane;
    ar[c] = (q < NQ) ? aself4[q] : f4z;
  }

  const bool isOdd = (p & 1) == 1;   // odds run the second color phase
  float nbomPh = 0.f;

  for (int it = 0; it < K; ++it) {
    const float* hk = hyp + ((size_t)it * PA + p) * 5;
    const float h0 = fabsf(hk[0]);
    const float h1 = fabsf(hk[1]);
    const float h2 = fabsf(hk[2]);
    const float h3 = fabsf(hk[3]);
    const float h4 = fabsf(hk[4]);

    // ---- two color phases: evens update, then odds (reading updated evens) ----
    #pragma unroll
    for (int phase = 0; phase < 2; ++phase) {
      if ((p & 1) == phase) {
        // xta = X . a from registers; wave32 butterfly reduction
        float part = 0.f;
        #pragma unroll
        for (int c = 0; c < NC4; ++c) {
          if (c < 6 || lane < 4) {
            part = fmaf(Xr[c].x, ar[c].x, part);
            part = fmaf(Xr[c].y, ar[c].y, part);
            part = fmaf(Xr[c].z, ar[c].z, part);
            part = fmaf(Xr[c].w, ar[c].w, part);
          }
        }
        #pragma unroll
        for (int off = 16; off > 0; off >>= 1)
          part += __shfl_xor(part, off, 32);

        const float om    = om_lds[w];
        const float resid = part + om - yv;
        nbomPh            = om_lds[nb0 * ROWS + r] + om_lds[nb1 * ROWS + r];
        const float newom = om - h4 * (resid + h2 * om * 2.f + lamv * 2.f - h2 * nbomPh);

        #pragma unroll
        for (int c = 0; c < NC4; ++c) {
          if (c < 6 || lane < 4) {
            const int q = c * 32 + lane;
            const float4 n0 = anb0_4[q];
            const float4 n1 = anb1_4[q];
            float4 nb;
            nb.x = n0.x + n1.x; nb.y = n0.y + n1.y;
            nb.z = n0.z + n1.z; nb.w = n0.w + n1.w;
            nbr[c] = nb;
            float4 na;
            na.x = ar[c].x - h1 * (Xr[c].x * resid + h0 * ar[c].x * 2.f + 2.f * mur[c].x - h0 * nb.x);
            na.y = ar[c].y - h1 * (Xr[c].y * resid + h0 * ar[c].y * 2.f + 2.f * mur[c].y - h0 * nb.y);
            na.z = ar[c].z - h1 * (Xr[c].z * resid + h0 * ar[c].z * 2.f + 2.f * mur[c].z - h0 * nb.z);
            na.w = ar[c].w - h1 * (Xr[c].w * resid + h0 * ar[c].w * 2.f + 2.f * mur[c].w - h0 * nb.w);
            ar[c] = na;
            aself4[q] = na;           // publish for neighbors
          }
        }
        if (lane == 0) om_lds[w] = newom;
      }
      __syncthreads();
    }

    // ---- dual update (mu, lam), register-resident.
    // Odds: neighbor values (evens) are final after phase A == what phase B read -> reuse regs.
    // Evens: odds changed in phase B -> re-read from LDS.
    {
      const float nbomD = isOdd ? nbomPh
                                : (om_lds[nb0 * ROWS + r] + om_lds[nb1 * ROWS + r]);
      const float om = om_lds[w];
      #pragma unroll
      for (int c = 0; c < NC4; ++c) {
        if (c < 6 || lane < 4) {
          float4 nb;
          if (isOdd) {
            nb = nbr[c];
          } else {
            const int q = c * 32 + lane;
            const float4 n0 = anb0_4[q];
            const float4 n1 = anb1_4[q];
            nb.x = n0.x + n1.x; nb.y = n0.y + n1.y;
            nb.z = n0.z + n1.z; nb.w = n0.w + n1.w;
          }
          mur[c].x = fmaf(h3, 2.f * ar[c].x - nb.x, mur[c].x);
          mur[c].y = fmaf(h3, 2.f * ar[c].y - nb.y, mur[c].y);
          mur[c].z = fmaf(h3, 2.f * ar[c].z - nb.z, mur[c].z);
          mur[c].w = fmaf(h3, 2.f * ar[c].w - nb.w, mur[c].w);
        }
      }
      lamv = fmaf(h4, 2.f * om - nbomD, lamv);
    }
    // No barrier needed here: dual reads only odd-agent LDS data (by evens),
    // which is next rewritten in phase B of the next iteration, i.e. after the
    // phase-A barrier that every wave must pass first.
  }

  __syncthreads();   // a_lds final for all agents before TDM store

  // ---- outputs: omega scalars, then a tiles back via TDM store ----
  if (lane == 0) out[(size_t)PA * NN * DD + pn] = om_lds[w];

  if (w == 0) {
    const unsigned lds_base = (unsigned)(size_t)(&a_lds[0][0]);
    #pragma unroll
    for (int pp = 0; pp < PA; ++pp) {
      u32x4 g0; i32x8 g1;
      i32x4 z4 = {0, 0, 0, 0};
      i32x8 z8 = {0, 0, 0, 0, 0, 0, 0, 0};
      unsigned long long ga =
          (unsigned long long)(size_t)(out + ((size_t)pp * NN + (size_t)blockIdx.x * ROWS) * DD);
      tdm_desc(ga, lds_base + (unsigned)(pp * ROWS * DD * 4), g0, g1);
      __builtin_amdgcn_tensor_store_from_lds(g0, g1, z4, z4, z8, 0);
    }
    __builtin_amdgcn_s_wait_tensorcnt(0);
  }
}

extern "C" void kernel_launch(void* const* d_in, const int* in_sizes, int n_in,
                              void* d_out, int out_size, void* d_ws, size_t ws_size,
                              hipStream_t stream) {
  (void)in_sizes; (void)n_in; (void)out_size; (void)d_ws; (void)ws_size;
  const float* X    = (const float*)d_in[0];
  const float* y    = (const float*)d_in[1];
  const float* a0   = (const float*)d_in[2];
  const float* om0  = (const float*)d_in[3];
  const float* mu0  = (const float*)d_in[4];
  const float* lam0 = (const float*)d_in[5];
  const float* hyp  = (const float*)d_in[6];
  const int*   K    = (const int*)d_in[7];
  float* out = (float*)d_out;

  dim3 grid(NN / ROWS);   // 1024 workgroups, each owning 2 samples x 4 agents
  DADMM_54640573940346_kernel<<<grid, BLOCK, 0, stream>>>(X, y, a0, om0, mu0, lam0, hyp, K, out);
}